// CartesianToJacobi_58995670777992
// MI455X (gfx1250) — compile-verified
//
#include <hip/hip_runtime.h>

// Cartesian -> Jacobi coordinate transform, B=524288 independent 11x11(x3) problems.
// Memory-bound (~323 MB total traffic -> ~14us floor @ 23.3 TB/s). WMMA is
// intentionally NOT used: per-batch matmul is 11x11x3 and the transform reduces
// to a prefix-FMA sweep; matrix cores cannot lower the HBM byte floor here.
// gfx1250-specific path used instead: async global->LDS staging (ASYNCcnt).

#define G_CONST 0.0002959122082855911f

#ifndef __has_builtin
#define __has_builtin(x) 0
#endif

#if __has_builtin(__builtin_amdgcn_global_load_async_to_lds_b128) && \
    __has_builtin(__builtin_amdgcn_s_wait_asynccnt)
#define USE_ASYNC_LDS 1
#else
#define USE_ASYNC_LDS 0
#endif

typedef float float4v __attribute__((ext_vector_type(4)));
typedef int   v4i     __attribute__((ext_vector_type(4)));

constexpr int N   = 11;   // bodies
constexpr int D   = 3;    // dims
constexpr int ND  = 33;   // N*D
constexpr int TPB = 128;  // batches (=threads) per block

#if USE_ASYNC_LDS
// Builtin signature (from clang diagnostic): (v4i addrspace(1)*, v4i addrspace(3)*, imm, imm)
#define ASYNC_COPY_B128(gsrc, ldsdst)                                   \
  __builtin_amdgcn_global_load_async_to_lds_b128(                       \
      (__attribute__((address_space(1))) v4i*)(gsrc),                   \
      (__attribute__((address_space(3))) v4i*)(ldsdst), 0, 0)
#endif

// Per-batch math: given m[0..10] and one column x[0..10], produce A*x.
// Row i>=1: x[i] - (sum_{j<i} m_j x_j) / M[i-1];  Row 0: (sum_j m_j x_j)/M_tot.
__device__ __forceinline__ void jacobi_column(const float* __restrict__ mr,
                                              const float* __restrict__ invM,
                                              const float* __restrict__ x,
                                              float* __restrict__ o) {
  float P = 0.0f;
#pragma unroll
  for (int i = 0; i < N; ++i) {
    if (i > 0) o[i] = fmaf(-P, invM[i - 1], x[i]);
    P = fmaf(mr[i], x[i], P);
  }
  o[0] = P * invM[N - 1];
}

__global__ __launch_bounds__(TPB) void jacobi_tile_kernel(
    const float* __restrict__ m, const float* __restrict__ q,
    const float* __restrict__ v, float* __restrict__ qj,
    float* __restrict__ vj, float* __restrict__ mu) {
  __shared__ alignas(16) float m_lds[TPB * N];   //  5.5 KB
  __shared__ alignas(16) float q_lds[TPB * ND];  // 16.5 KB
  __shared__ alignas(16) float v_lds[TPB * ND];  // 16.5 KB

  const int tid = threadIdx.x;
  const long long b0 = (long long)blockIdx.x * TPB;

  const float4v* mg = (const float4v*)(m + b0 * N);
  const float4v* qg = (const float4v*)(q + b0 * ND);
  const float4v* vg = (const float4v*)(v + b0 * ND);
  float4v* ml4 = (float4v*)m_lds;
  float4v* ql4 = (float4v*)q_lds;
  float4v* vl4 = (float4v*)v_lds;

  constexpr int M4 = TPB * N / 4;   // 352 float4s
  constexpr int Q4 = TPB * ND / 4;  // 1056 float4s

  // ---- Stage the block's tile into LDS with coalesced b128 transfers ----
#if USE_ASYNC_LDS
#pragma unroll 1
  for (int k = tid; k < M4; k += TPB) ASYNC_COPY_B128(mg + k, ml4 + k);
#pragma unroll 1
  for (int k = tid; k < Q4; k += TPB) ASYNC_COPY_B128(qg + k, ql4 + k);
#pragma unroll 1
  for (int k = tid; k < Q4; k += TPB) ASYNC_COPY_B128(vg + k, vl4 + k);
  __builtin_amdgcn_s_wait_asynccnt(0);
#else
#pragma unroll 1
  for (int k = tid; k < M4; k += TPB) ml4[k] = __builtin_nontemporal_load(mg + k);
#pragma unroll 1
  for (int k = tid; k < Q4; k += TPB) ql4[k] = __builtin_nontemporal_load(qg + k);
#pragma unroll 1
  for (int k = tid; k < Q4; k += TPB) vl4[k] = __builtin_nontemporal_load(vg + k);
#endif
  __syncthreads();

  // ---- Per-thread batch compute (strides 11/33 dwords -> bank-conflict-free) ----
  float mr[N], Ms[N], invM[N];
#pragma unroll
  for (int j = 0; j < N; ++j) mr[j] = m_lds[tid * N + j];
  Ms[0] = mr[0];
#pragma unroll
  for (int j = 1; j < N; ++j) Ms[j] = Ms[j - 1] + mr[j];
#pragma unroll
  for (int j = 0; j < N; ++j) invM[j] = 1.0f / Ms[j];
  // mu overwrites the m tile in place (only this thread reads these slots now)
#pragma unroll
  for (int j = 0; j < N; ++j) m_lds[tid * N + j] = G_CONST * Ms[j];

#pragma unroll
  for (int c = 0; c < 2 * D; ++c) {
    float* buf = (c < D) ? q_lds : v_lds;
    const int d = (c < D) ? c : c - D;
    float x[N], o[N];
#pragma unroll
    for (int i = 0; i < N; ++i) x[i] = buf[tid * ND + i * D + d];
    jacobi_column(mr, invM, x, o);
#pragma unroll
    for (int i = 0; i < N; ++i) buf[tid * ND + i * D + d] = o[i];
  }
  __syncthreads();

  // ---- Stream results out: coalesced non-temporal b128 stores ----
  float4v* qo = (float4v*)(qj + b0 * ND);
  float4v* vo = (float4v*)(vj + b0 * ND);
  float4v* uo = (float4v*)(mu + b0 * N);
#pragma unroll 1
  for (int k = tid; k < Q4; k += TPB) __builtin_nontemporal_store(ql4[k], qo + k);
#pragma unroll 1
  for (int k = tid; k < Q4; k += TPB) __builtin_nontemporal_store(vl4[k], vo + k);
#pragma unroll 1
  for (int k = tid; k < M4; k += TPB) __builtin_nontemporal_store(ml4[k], uo + k);
}

// Scalar tail for B % TPB (not used for B=524288, kept for generality).
__global__ void jacobi_tail_kernel(const float* __restrict__ m,
                                   const float* __restrict__ q,
                                   const float* __restrict__ v,
                                   float* __restrict__ qj, float* __restrict__ vj,
                                   float* __restrict__ mu, long long Bstart,
                                   long long Btotal) {
  long long b = Bstart + (long long)blockIdx.x * blockDim.x + threadIdx.x;
  if (b >= Btotal) return;
  float mr[N], Ms[N], invM[N];
#pragma unroll
  for (int j = 0; j < N; ++j) mr[j] = m[b * N + j];
  Ms[0] = mr[0];
#pragma unroll
  for (int j = 1; j < N; ++j) Ms[j] = Ms[j - 1] + mr[j];
#pragma unroll
  for (int j = 0; j < N; ++j) invM[j] = 1.0f / Ms[j];
#pragma unroll
  for (int j = 0; j < N; ++j) mu[b * N + j] = G_CONST * Ms[j];
#pragma unroll
  for (int c = 0; c < 2 * D; ++c) {
    const float* src = (c < D) ? q : v;
    float* dst = (c < D) ? qj : vj;
    const int d = (c < D) ? c : c - D;
    float x[N], o[N];
#pragma unroll
    for (int i = 0; i < N; ++i) x[i] = src[b * ND + i * D + d];
    jacobi_column(mr, invM, x, o);
#pragma unroll
    for (int i = 0; i < N; ++i) dst[b * ND + i * D + d] = o[i];
  }
}

extern "C" void kernel_launch(void* const* d_in, const int* in_sizes, int n_in,
                              void* d_out, int out_size, void* d_ws, size_t ws_size,
                              hipStream_t stream) {
  (void)n_in; (void)d_ws; (void)ws_size; (void)out_size;
  const float* m = (const float*)d_in[0];
  const float* q = (const float*)d_in[1];
  const float* v = (const float*)d_in[2];
  const long long B = (long long)in_sizes[0] / N;  // 524288

  float* qj = (float*)d_out;                 // [B, 11, 3]
  float* vj = qj + (size_t)B * ND;           // [B, 11, 3]
  float* mu = vj + (size_t)B * ND;           // [B, 11]

  const long long nfull = B / TPB;
  if (nfull > 0) {
    jacobi_tile_kernel<<<(unsigned)nfull, TPB, 0, stream>>>(m, q, v, qj, vj, mu);
  }
  const long long rem = B % TPB;
  if (rem > 0) {
    const long long start = nfull * (long long)TPB;
    jacobi_tail_kernel<<<(unsigned)((rem + 127) / 128), 128, 0, stream>>>(
        m, q, v, qj, vj, mu, start, B);
  }
}